// MemoryBlock_6828998001203
// MI455X (gfx1250) — compile-verified
//
#include <hip/hip_runtime.h>

typedef float v2f __attribute__((ext_vector_type(2)));
typedef float v4f __attribute__((ext_vector_type(4)));
typedef float v8f __attribute__((ext_vector_type(8)));

#define KQ   50000
#define CC   64     // channels
#define EKD  128    // query dim
#define EID  256    // inter dim
#define ROWS 16     // b-rows per block

__global__ __launch_bounds__(256) void memblock_fused(
    const int*   __restrict__ x,
    const float* __restrict__ value_memory,
    const float* __restrict__ W_q,
    const float* __restrict__ W_i,
    const float* __restrict__ key_memory,
    float*       __restrict__ out)
{
    __shared__ float s_logits[ROWS * CC];                                 // 4 KB (becomes weights)
    __shared__ __attribute__((aligned(16))) float s_inter[ROWS * EID];    // 16 KB
    __shared__ int   s_xid[ROWS];
    __shared__ int   s_qid[ROWS];

    const int tid = threadIdx.x;
    const int b0  = blockIdx.x * ROWS;

    // ---- Phase 0: indices (replicate Python modulo: x==0 -> qid 50000) ----
    if (tid < ROWS) {
        int xid = x[b0 + tid];
        s_xid[tid] = xid;
        s_qid[tid] = ((xid - 1) % KQ + KQ) % KQ + 1;
    }
    __syncthreads();

    // ---- Phase 1: logits[16x64] = q @ key^T via V_WMMA_F32_16X16X4_F32 (wave 0) ----
    if (tid < 32) {
        const int lane  = tid;
        const int m16   = lane & 15;   // M row (A) / N col (B) within tile
        const int khalf = lane >> 4;   // K pair selector: 0 -> K{0,1}, 1 -> K{2,3}
        const float* qrow = W_q + (size_t)s_qid[m16] * EKD;

        v8f acc0 = {}, acc1 = {}, acc2 = {}, acc3 = {};
        #pragma unroll 4
        for (int kk = 0; kk < EKD / 4; ++kk) {
            const int koff = kk * 4 + khalf * 2;
            // A: 16x4 f32 tile, row = m16, K pair per lane-half
            v2f a = *(const v2f*)(qrow + koff);
            // B: 4x16 f32 tiles, B[k][n] = key_memory[n][k] (einsum 'bk,ck->bc')
            v2f bf0 = *(const v2f*)(key_memory + (m16 +  0) * EKD + koff);
            v2f bf1 = *(const v2f*)(key_memory + (m16 + 16) * EKD + koff);
            v2f bf2 = *(const v2f*)(key_memory + (m16 + 32) * EKD + koff);
            v2f bf3 = *(const v2f*)(key_memory + (m16 + 48) * EKD + koff);
            acc0 = __builtin_amdgcn_wmma_f32_16x16x4_f32(false, a, false, bf0, (short)0, acc0, false, false);
            acc1 = __builtin_amdgcn_wmma_f32_16x16x4_f32(false, a, false, bf1, (short)0, acc1, false, false);
            acc2 = __builtin_amdgcn_wmma_f32_16x16x4_f32(false, a, false, bf2, (short)0, acc2, false, false);
            acc3 = __builtin_amdgcn_wmma_f32_16x16x4_f32(false, a, false, bf3, (short)0, acc3, false, false);
        }
        // C/D layout: VGPR v, lanes 0-15 -> M=v, lanes 16-31 -> M=v+8; N = lane&15
        #pragma unroll
        for (int v = 0; v < 8; ++v) {
            const int m = v + khalf * 8;
            s_logits[m * CC + m16 +  0] = acc0[v];
            s_logits[m * CC + m16 + 16] = acc1[v];
            s_logits[m * CC + m16 + 32] = acc2[v];
            s_logits[m * CC + m16 + 48] = acc3[v];
        }
    }

    // ---- Phase 1b: inter = tanh(W_i[xid]) for all 16 rows (all 256 threads) ----
    #pragma unroll
    for (int r = 0; r < ROWS; ++r) {
        const float* wrow = W_i + (size_t)s_xid[r] * EID;
        s_inter[r * EID + tid] = tanhf(wrow[tid]);
    }
    __syncthreads();

    // ---- Phase 2: softmax over the 64 channels, in place (thread-per-row) ----
    if (tid < ROWS) {
        float* lrow = s_logits + tid * CC;
        float mx = lrow[0];
        #pragma unroll
        for (int c = 1; c < CC; ++c) mx = fmaxf(mx, lrow[c]);
        float sum = 0.f;
        #pragma unroll
        for (int c = 0; c < CC; ++c) { float e = __expf(lrow[c] - mx); lrow[c] = e; sum += e; }
        const float inv = 1.f / sum;
        #pragma unroll
        for (int c = 0; c < CC; ++c) lrow[c] *= inv;
    }
    __syncthreads();

    // ---- Phase 3: stream out = value + weight[b,c] * inter[b,e]  (float4, NT) ----
    const size_t base4 = (size_t)blockIdx.x * (ROWS * CC * EID / 4);
    const v4f* __restrict__ vin  = (const v4f*)value_memory + base4;
    v4f*       __restrict__ vout = (v4f*)out + base4;
    #pragma unroll 4
    for (int i = 0; i < (ROWS * CC * EID / 4) / 256; ++i) {
        const int idx4 = i * 256 + tid;          // 0..65535, coalesced
        const int e4   = idx4 & (EID / 4 - 1);   // 0..63
        const int c    = (idx4 >> 6) & (CC - 1); // 0..63
        const int r    = idx4 >> 12;             // 0..15
        const float w  = s_logits[r * CC + c];
        const v4f itv  = *(const v4f*)(s_inter + r * EID + e4 * 4);
        v4f vm = __builtin_nontemporal_load(vin + idx4);
        v4f o  = vm + w * itv;
        __builtin_nontemporal_store(o, vout + idx4);
    }
}

extern "C" void kernel_launch(void* const* d_in, const int* in_sizes, int n_in,
                              void* d_out, int out_size, void* d_ws, size_t ws_size,
                              hipStream_t stream) {
    const int*   x     = (const int*)d_in[0];
    const float* value = (const float*)d_in[1];
    const float* W_q   = (const float*)d_in[2];
    const float* W_i   = (const float*)d_in[3];
    const float* keym  = (const float*)d_in[4];
    float*       out   = (float*)d_out;

    const int B = in_sizes[0];              // 16384
    const int nblocks = B / ROWS;           // 1024
    memblock_fused<<<nblocks, 256, 0, stream>>>(x, value, W_q, W_i, keym, out);
}